// GAT_54116587929911
// MI455X (gfx1250) — compile-verified
//
#include <hip/hip_runtime.h>
#include <math.h>

#define N_NODES 100000
#define N_EDGES 3200000
#define F 128
#define NHEAD 4
#define DH 32
#define NEG_SLOPE 0.2f

typedef float v2f __attribute__((ext_vector_type(2)));
typedef float v8f __attribute__((ext_vector_type(8)));
typedef int   v4i __attribute__((ext_vector_type(4)));

#if __has_builtin(__builtin_amdgcn_global_load_async_to_lds_b128)
#define HAVE_ASYNC_LDS 1
#else
#define HAVE_ASYNC_LDS 0
#endif

__device__ __forceinline__ void wait_async0() {
#if __has_builtin(__builtin_amdgcn_s_wait_asynccnt)
    __builtin_amdgcn_s_wait_asynccnt(0);
#else
    asm volatile("s_wait_asynccnt 0" ::: "memory");
#endif
}

// ---- order-preserving float <-> uint encoding for atomic max ----
__device__ __forceinline__ unsigned enc_f(float f) {
    unsigned u = __float_as_uint(f);
    return (u & 0x80000000u) ? ~u : (u | 0x80000000u);
}
__device__ __forceinline__ float dec_f(unsigned e) {
    unsigned u = (e & 0x80000000u) ? (e & 0x7FFFFFFFu) : ~e;
    return __uint_as_float(u);
}
__device__ __forceinline__ float lrelu(float x) {
    return x > 0.f ? x : NEG_SLOPE * x;
}

// ---- init: m = enc(-inf), s = 0, acc = 0 ----
__global__ void k_init(unsigned* __restrict__ m_enc, float* __restrict__ s,
                       float* __restrict__ acc) {
    int i = blockIdx.x * blockDim.x + threadIdx.x;
    if (i < N_NODES * NHEAD) { m_enc[i] = 0x007FFFFFu; s[i] = 0.f; }
    if (i < N_NODES * F) acc[i] = 0.f;
}

// ---- pack W into K-pair float2 layout: Bp[(k/2)*256 + n*2 + (k&1)] = B[k*128+n]
__global__ void k_pack_b(const float* __restrict__ B, float* __restrict__ Bp) {
    int i = blockIdx.x * blockDim.x + threadIdx.x;
    if (i >= F * F) return;
    int k = i >> 7, n = i & 127;
    Bp[((k >> 1) << 8) + (n << 1) + (k & 1)] = B[i];
}

// ---- GEMM: C[M,128] = A[M,128] @ B[128,128] via V_WMMA_F32_16X16X4_F32 ----
// One wave -> 16 rows x 128 cols. Pre-packed B image async-copied verbatim
// into LDS (ASYNCcnt path), fragments read as adjacent float2 K-pairs.
__global__ __launch_bounds__(256) void k_gemm_wmma(const float* __restrict__ A,
                                                   const float* __restrict__ Bp,
                                                   float* __restrict__ C) {
    __shared__ float Bs[F * F];  // 64 KB, verbatim copy of packed B image
    int tid = threadIdx.x;
#if HAVE_ASYNC_LDS
    // 16-byte chunks, 16 async copies per thread, direct to LDS (no VGPR bounce)
    for (int i = tid * 4; i < F * F; i += 256 * 4) {
        __builtin_amdgcn_global_load_async_to_lds_b128(
            (v4i*)(Bp + i), (v4i*)(Bs + i), 0, 0);
    }
    wait_async0();   // each wave drains its own ASYNCcnt before the barrier
#else
    for (int i = tid; i < F * F; i += 256) Bs[i] = Bp[i];
#endif
    __syncthreads();

    int wave = tid >> 5, lane = tid & 31;
    int m0 = (blockIdx.x * 8 + wave) * 16;
    if (m0 >= N_NODES) return;          // wave-uniform, EXEC stays all-ones

    int lh = lane >> 4;                 // 0 or 1: selects K pair
    int ll = lane & 15;                 // row (A) / col (B)
    v8f acc[8] = {};                    // 8 N-tiles of 16x16 f32

    const float* arow = A + (size_t)(m0 + ll) * F;
    const v2f* bs2 = (const v2f*)Bs;

    for (int kb = 0; kb < F; kb += 4) {
        // A 16x4 fragment: lane<16 holds {K=kb,kb+1}, lane>=16 holds {K=kb+2,kb+3}
        v2f a = *(const v2f*)(arow + kb + (lh << 1));
        int kh = ((kb >> 1) + lh) << 7;  // packed pair-row base
#pragma unroll
        for (int t = 0; t < 8; ++t) {
            // B 4x16 fragment: {B[k0][n], B[k0+1][n]} adjacent in packed LDS
            v2f b = bs2[kh + t * 16 + ll];
            acc[t] = __builtin_amdgcn_wmma_f32_16x16x4_f32(
                false, a, false, b, (short)0, acc[t], false, false);
        }
    }
    // C layout: VGPR r -> M = m0 + r + 8*lh, N = t*16 + ll
#pragma unroll
    for (int t = 0; t < 8; ++t) {
        float* crow = C + (size_t)(m0 + (lh << 3)) * F + t * 16 + ll;
#pragma unroll
        for (int r = 0; r < 8; ++r) crow[(size_t)r * F] = acc[t][r];
    }
}

// ---- per-(node,head) attention scores el/er ----
__global__ void k_attn(const float* __restrict__ feat, const float* __restrict__ al,
                       const float* __restrict__ ar, float* __restrict__ el,
                       float* __restrict__ er) {
    int i = blockIdx.x * blockDim.x + threadIdx.x;
    if (i >= N_NODES * NHEAD) return;
    int n = i >> 2, h = i & 3;
    const float4* fp  = (const float4*)(feat + (size_t)n * F + h * DH);
    const float4* alp = (const float4*)(al + h * DH);
    const float4* arp = (const float4*)(ar + h * DH);
    float sl = 0.f, sr = 0.f;
#pragma unroll
    for (int d = 0; d < DH / 4; ++d) {
        float4 f = fp[d], a = alp[d], b = arp[d];
        sl += f.x * a.x + f.y * a.y + f.z * a.z + f.w * a.w;
        sr += f.x * b.x + f.y * b.y + f.z * b.z + f.w * b.w;
    }
    el[i] = sl; er[i] = sr;
}

// ---- edge pass 1: segment max over dst (b128 gathers of 4 heads) ----
__global__ void k_edge_max(const int* __restrict__ src, const int* __restrict__ dst,
                           const float* __restrict__ el, const float* __restrict__ er,
                           unsigned* __restrict__ m_enc) {
    int e = blockIdx.x * blockDim.x + threadIdx.x;
    if (e >= N_EDGES) return;
    int s = src[e], d = dst[e];
    float4 a = *(const float4*)(el + 4 * (size_t)s);
    float4 b = *(const float4*)(er + 4 * (size_t)d);
    unsigned* mp = m_enc + 4 * (size_t)d;
    atomicMax(mp + 0, enc_f(lrelu(a.x + b.x)));
    atomicMax(mp + 1, enc_f(lrelu(a.y + b.y)));
    atomicMax(mp + 2, enc_f(lrelu(a.z + b.z)));
    atomicMax(mp + 3, enc_f(lrelu(a.w + b.w)));
}

// ---- decode max; non-finite (no edges) -> 0; store float in place ----
__global__ void k_fin_max(unsigned* __restrict__ m_enc) {
    int i = blockIdx.x * blockDim.x + threadIdx.x;
    if (i >= N_NODES * NHEAD) return;
    float m = dec_f(m_enc[i]);
    if (!isfinite(m)) m = 0.f;
    ((float*)m_enc)[i] = m;
}

// ---- edge pass 2: segment sum of exp(e - m[dst]) ----
__global__ void k_edge_sum(const int* __restrict__ src, const int* __restrict__ dst,
                           const float* __restrict__ el, const float* __restrict__ er,
                           const float* __restrict__ m, float* __restrict__ s_out) {
    int e = blockIdx.x * blockDim.x + threadIdx.x;
    if (e >= N_EDGES) return;
    int s = src[e], d = dst[e];
    float4 a  = *(const float4*)(el + 4 * (size_t)s);
    float4 b  = *(const float4*)(er + 4 * (size_t)d);
    float4 mm = *(const float4*)(m  + 4 * (size_t)d);
    float* sp = s_out + 4 * (size_t)d;
    atomicAdd(sp + 0, expf(lrelu(a.x + b.x) - mm.x));
    atomicAdd(sp + 1, expf(lrelu(a.y + b.y) - mm.y));
    atomicAdd(sp + 2, expf(lrelu(a.z + b.z) - mm.z));
    atomicAdd(sp + 3, expf(lrelu(a.w + b.w) - mm.w));
}

// ---- s -> 1/max(s, 1e-20) in place ----
__global__ void k_fin_sum(float* __restrict__ s) {
    int i = blockIdx.x * blockDim.x + threadIdx.x;
    if (i >= N_NODES * NHEAD) return;
    s[i] = 1.f / fmaxf(s[i], 1e-20f);
}

// ---- edge pass 3: weighted message scatter. One wave per edge, 4 floats/lane ----
__global__ __launch_bounds__(256) void k_edge_scatter(
    const int* __restrict__ src, const int* __restrict__ dst,
    const float* __restrict__ ew, const float* __restrict__ el,
    const float* __restrict__ er, const float* __restrict__ m,
    const float* __restrict__ inv_s, const float* __restrict__ feat,
    float* __restrict__ acc) {
    int e = blockIdx.x * 8 + (threadIdx.x >> 5);
    if (e >= N_EDGES) return;          // wave-uniform
    int lane = threadIdx.x & 31;
    int h = lane >> 3;                 // head for this lane's 4 features
    int s = src[e], d = dst[e];
    float x = lrelu(el[s * NHEAD + h] + er[d * NHEAD + h]);
    float c = expf(x - m[d * NHEAD + h]) * inv_s[d * NHEAD + h] * ew[e];
    int f0 = lane << 2;
    float4 f = *(const float4*)(feat + (size_t)s * F + f0);
    float* o = acc + (size_t)d * F + f0;
    atomicAdd(o + 0, c * f.x);
    atomicAdd(o + 1, c * f.y);
    atomicAdd(o + 2, c * f.z);
    atomicAdd(o + 3, c * f.w);
}

// ---- bias + ELU in place ----
__global__ void k_bias_elu(float* __restrict__ acc, const float* __restrict__ bias) {
    int i = blockIdx.x * blockDim.x + threadIdx.x;
    if (i >= N_NODES * F) return;
    float x = acc[i] + bias[i & 127];
    acc[i] = x > 0.f ? x : expf(x) - 1.f;
}

extern "C" void kernel_launch(void* const* d_in, const int* in_sizes, int n_in,
                              void* d_out, int out_size, void* d_ws, size_t ws_size,
                              hipStream_t stream) {
    const float* in_feat = (const float*)d_in[0];
    const float* ew      = (const float*)d_in[1];
    const int*   src     = (const int*)d_in[2];
    const int*   dst     = (const int*)d_in[3];
    const float* W0      = (const float*)d_in[4];
    const float* al0     = (const float*)d_in[5];
    const float* ar0     = (const float*)d_in[6];
    const float* b0      = (const float*)d_in[7];
    const float* W1      = (const float*)d_in[8];
    const float* al1     = (const float*)d_in[9];
    const float* ar1     = (const float*)d_in[10];
    const float* b1      = (const float*)d_in[11];
    float* out = (float*)d_out;

    char* ws = (char*)d_ws;
    const size_t NF = (size_t)N_NODES * F * sizeof(float);      // 51.2 MB
    const size_t NH = (size_t)N_NODES * NHEAD * sizeof(float);  // 1.6 MB
    float*    feat = (float*)(ws);
    float*    h1   = (float*)(ws + NF);
    float*    el   = (float*)(ws + 2 * NF);
    float*    er   = (float*)(ws + 2 * NF + NH);
    unsigned* m    = (unsigned*)(ws + 2 * NF + 2 * NH);
    float*    s    = (float*)(ws + 2 * NF + 3 * NH);
    float*    Bp   = (float*)(ws + 2 * NF + 4 * NH);            // 64 KB packed W

    const int B256   = 256;
    const int gNF    = (N_NODES * F + B256 - 1) / B256;
    const int gNH    = (N_NODES * NHEAD + B256 - 1) / B256;
    const int gE     = (N_EDGES + B256 - 1) / B256;
    const int gEwave = (N_EDGES + 7) / 8;
    const int gGemm  = (N_NODES / 16 + 7) / 8;  // 6250 M-tiles, 8 per block
    const int gPack  = (F * F + B256 - 1) / B256;

    auto layer = [&](const float* hin, const float* W, const float* al,
                     const float* ar, const float* bias, float* accout) {
        k_init<<<gNF, B256, 0, stream>>>(m, s, accout);
        k_pack_b<<<gPack, B256, 0, stream>>>(W, Bp);
        k_gemm_wmma<<<gGemm, B256, 0, stream>>>(hin, Bp, feat);
        k_attn<<<gNH, B256, 0, stream>>>(feat, al, ar, el, er);
        k_edge_max<<<gE, B256, 0, stream>>>(src, dst, el, er, m);
        k_fin_max<<<gNH, B256, 0, stream>>>(m);
        k_edge_sum<<<gE, B256, 0, stream>>>(src, dst, el, er, (const float*)m, s);
        k_fin_sum<<<gNH, B256, 0, stream>>>(s);
        k_edge_scatter<<<gEwave, B256, 0, stream>>>(src, dst, ew, el, er,
                                                    (const float*)m, s, feat, accout);
        k_bias_elu<<<gNF, B256, 0, stream>>>(accout, bias);
    };

    layer(in_feat, W0, al0, ar0, b0, h1);
    layer(h1, W1, al1, ar1, b1, out);
}